// RMAC_24816321036475
// MI455X (gfx1250) — compile-verified
//
#include <hip/hip_runtime.h>
#include <math.h>
#include <stdint.h>

#define RMAC_B      32
#define RMAC_C      2048
#define RMAC_TOTAL  65536      // B*C
#define RMAC_R      14         // distinct regions: full(w=2), 4x 10x10, 9x 8x8
#define RMAC_GROUPS (RMAC_R * RMAC_B)   // 448 (region,b) norm groups
#define NEGMAX      (-3.402823466e+38f)

#define TILES_PER_BLK 64                       // 64 KiB of input per block
#define TILE_F4       64                       // 256 floats = 64 float4 per tile
#define TILE_F4_PAD   65                       // +16B pad -> bank rotation
#define LDS_BYTES     (TILES_PER_BLK * TILE_F4_PAD * 16)   // 66560 B

typedef float v2f __attribute__((ext_vector_type(2)));
typedef float v8f __attribute__((ext_vector_type(8)));

// ---------------------------------------------------------------------------
// Kernel 1: per-(b,c) spatial max for all 14 regions.
// CDNA5 data path: stage 64 tiles (64 KiB) per block into LDS with
// GLOBAL_LOAD_ASYNC_TO_LDS_B128 (coalesced 512B/wave/instruction, tracked by
// ASYNCcnt), s_wait_asynccnt + barrier, then each thread max-reduces its own
// padded tile out of LDS. Output: ws_m[r * 65536 + tid].
// ---------------------------------------------------------------------------
__global__ __launch_bounds__(64) void rmac_pool(const float* __restrict__ x,
                                                float* __restrict__ m) {
  extern __shared__ float4 tile_lds[];   // [TILES_PER_BLK][TILE_F4_PAD]
  const int t   = threadIdx.x;           // 0..63
  const int blk = blockIdx.x;            // 1024 blocks
  const float* gbase = x + (size_t)blk * TILES_PER_BLK * 256;

  // ---- async stage: iteration i copies tile i; thread t moves element t ----
#pragma unroll
  for (int i = 0; i < TILES_PER_BLK; ++i) {
    unsigned lds_off =
        (unsigned)(uintptr_t)(&tile_lds[i * TILE_F4_PAD + t]); // LDS byte addr
    unsigned voff = (unsigned)(i * 1024 + t * 16);             // global byte off
    asm volatile("global_load_async_to_lds_b128 %0, %1, %2"
                 :
                 : "v"(lds_off), "v"(voff), "s"(gbase)
                 : "memory");
  }
  asm volatile("s_wait_asynccnt 0x0" ::: "memory");
  __syncthreads();

  // ---- per-thread region maxes from LDS tile t ----
  const float4* tp = &tile_lds[t * TILE_F4_PAD];

  float mfull  = NEGMAX;
  float m10_00 = NEGMAX, m10_01 = NEGMAX, m10_10 = NEGMAX, m10_11 = NEGMAX;
  float m8[3][3];
#pragma unroll
  for (int i = 0; i < 3; ++i)
#pragma unroll
    for (int j = 0; j < 3; ++j) m8[i][j] = NEGMAX;

#pragma unroll
  for (int r = 0; r < 16; ++r) {
    float4 q0 = tp[r * 4 + 0];
    float4 q1 = tp[r * 4 + 1];
    float4 q2 = tp[r * 4 + 2];
    float4 q3 = tp[r * 4 + 3];
    float row[16] = {q0.x, q0.y, q0.z, q0.w, q1.x, q1.y, q1.z, q1.w,
                     q2.x, q2.y, q2.z, q2.w, q3.x, q3.y, q3.z, q3.w};
    // 8-wide column-segment maxes (cols 0-7, 4-11, 8-15)
    float c8_0 = row[0], c8_1 = row[4], c8_2 = row[8];
#pragma unroll
    for (int k = 1; k < 8; ++k) {
      c8_0 = fmaxf(c8_0, row[k]);
      c8_1 = fmaxf(c8_1, row[4 + k]);
      c8_2 = fmaxf(c8_2, row[8 + k]);
    }
    float cfull = fmaxf(c8_0, c8_2);                     // cols 0-15
    float c10_0 = fmaxf(c8_0, fmaxf(row[8], row[9]));    // cols 0-9
    float c10_1 = fmaxf(c8_2, fmaxf(row[6], row[7]));    // cols 6-15

    mfull = fmaxf(mfull, cfull);
    if (r < 10) {                       // rows 0-9
      m10_00 = fmaxf(m10_00, c10_0);
      m10_01 = fmaxf(m10_01, c10_1);
    }
    if (r >= 6) {                       // rows 6-15
      m10_10 = fmaxf(m10_10, c10_0);
      m10_11 = fmaxf(m10_11, c10_1);
    }
    if (r < 8) {                        // rows 0-7
      m8[0][0] = fmaxf(m8[0][0], c8_0);
      m8[0][1] = fmaxf(m8[0][1], c8_1);
      m8[0][2] = fmaxf(m8[0][2], c8_2);
    }
    if (r >= 4 && r < 12) {             // rows 4-11
      m8[1][0] = fmaxf(m8[1][0], c8_0);
      m8[1][1] = fmaxf(m8[1][1], c8_1);
      m8[1][2] = fmaxf(m8[1][2], c8_2);
    }
    if (r >= 8) {                       // rows 8-15
      m8[2][0] = fmaxf(m8[2][0], c8_0);
      m8[2][1] = fmaxf(m8[2][1], c8_1);
      m8[2][2] = fmaxf(m8[2][2], c8_2);
    }
  }

  const int tid = blk * TILES_PER_BLK + t;   // global (b,c) index
  m[0 * RMAC_TOTAL + tid] = mfull;
  m[1 * RMAC_TOTAL + tid] = m10_00;
  m[2 * RMAC_TOTAL + tid] = m10_01;
  m[3 * RMAC_TOTAL + tid] = m10_10;
  m[4 * RMAC_TOTAL + tid] = m10_11;
#pragma unroll
  for (int i = 0; i < 3; ++i)
#pragma unroll
    for (int j = 0; j < 3; ++j)
      m[(5 + i * 3 + j) * RMAC_TOTAL + tid] = m8[i][j];
}

// ---------------------------------------------------------------------------
// Kernel 2: per-(region,b) inverse L2 norm over C=2048 channels.
// One wave handles 16 groups; two fp32 partial sums per lane map exactly to
// the A-matrix 16x4 layout; one exact-fp32 V_WMMA_F32_16X16X4_F32 against a
// ones B-matrix performs the cross-lane reduction.
// ---------------------------------------------------------------------------
__global__ __launch_bounds__(32) void rmac_norm(const float* __restrict__ m,
                                                float* __restrict__ invn) {
  int lane = threadIdx.x;          // 0..31, one full wave (EXEC all ones)
  int g0 = blockIdx.x * 16;        // first group of this wave
  int g = g0 + (lane & 15);        // group = r*32 + b
  int h = lane >> 4;               // which channel half

  const float4* p =
      reinterpret_cast<const float4*>(m + (size_t)g * RMAC_C + h * (RMAC_C / 2));
  float acc0 = 0.0f, acc1 = 0.0f;
#pragma unroll 4
  for (int i = 0; i < RMAC_C / 8; ++i) {   // 256 float4 loads per lane
    float4 v = p[i];
    acc0 += v.x * v.x + v.y * v.y;
    acc1 += v.z * v.z + v.w * v.w;
  }

  v2f a;  a.x = acc0; a.y = acc1;
  v2f b;  b.x = 1.0f; b.y = 1.0f;          // ones B-matrix: layout-agnostic
  v8f c = {};
  c = __builtin_amdgcn_wmma_f32_16x16x4_f32(
      /*neg_a=*/false, a, /*neg_b=*/false, b,
      /*c_mod=*/(short)0, c, /*reuse_a=*/false, /*reuse_b=*/false);

  // D layout: VGPR v holds M=v (lanes 0-15) and M=v+8 (lanes 16-31).
  if (lane == 0) {
#pragma unroll
    for (int v = 0; v < 8; ++v)
      invn[g0 + v] = 1.0f / (sqrtf(c[v]) + 1e-6f);
  } else if (lane == 16) {
#pragma unroll
    for (int v = 0; v < 8; ++v)
      invn[g0 + 8 + v] = 1.0f / (sqrtf(c[v]) + 1e-6f);
  }
}

// ---------------------------------------------------------------------------
// Kernel 3: out[b,c] = sum_r w_r * m[r,b,c] * invn[r,b]  (w_0 = 2: the l=1
// region grid entry duplicates the initial full-image MAC).
// ---------------------------------------------------------------------------
__global__ __launch_bounds__(256) void rmac_combine(const float* __restrict__ m,
                                                    const float* __restrict__ invn,
                                                    float* __restrict__ out) {
  int tid = blockIdx.x * blockDim.x + threadIdx.x;  // b*2048 + c
  int b = tid >> 11;
  float sum = 0.0f;
#pragma unroll
  for (int r = 0; r < RMAC_R; ++r) {
    float val = m[(size_t)r * RMAC_TOTAL + tid];
    float w = (r == 0) ? 2.0f : 1.0f;
    sum += w * val * invn[r * RMAC_B + b];
  }
  out[tid] = sum;
}

// ---------------------------------------------------------------------------
extern "C" void kernel_launch(void* const* d_in, const int* in_sizes, int n_in,
                              void* d_out, int out_size, void* d_ws, size_t ws_size,
                              hipStream_t stream) {
  const float* x = (const float*)d_in[0];
  float* out = (float*)d_out;

  float* ws_m   = (float*)d_ws;                       // [14][65536] region maxes
  float* ws_inv = ws_m + (size_t)RMAC_R * RMAC_TOTAL; // [448] inverse norms

  rmac_pool<<<RMAC_TOTAL / TILES_PER_BLK, 64, LDS_BYTES, stream>>>(x, ws_m);
  rmac_norm<<<RMAC_GROUPS / 16, 32, 0, stream>>>(ws_m, ws_inv);
  rmac_combine<<<RMAC_TOTAL / 256, 256, 0, stream>>>(ws_m, ws_inv, out);
}